// SPD_29008209117456
// MI455X (gfx1250) — compile-verified
//
#include <hip/hip_runtime.h>
#include <math.h>

// SPD matrix log: out_b = V diag(log(max(w,eps))) V^T  for  h_b + eps*I = V diag(w) V^T
// One workgroup per 128x128 matrix. Parallel cyclic Jacobi eigensolver in LDS
// (CDNA5 320KB WGP LDS makes the 130KB A+V footprint legal), then fp32 WMMA
// (v_wmma_f32_16x16x4_f32) reconstruction.

typedef __attribute__((ext_vector_type(2))) float v2f;
typedef __attribute__((ext_vector_type(8))) float v8f;

#define N        128
#define LDA      129         // padded LDS row stride (129 % 64 banks == 1 -> conflict-free columns)
#define NPAIR    64
#define NSWEEPS  8
#define BLOCK    256
#define EPS      1e-4f
#define RECT_EPS 1e-4f

extern __shared__ float smem[];

__global__ __launch_bounds__(BLOCK) void spd_logm_kernel(const float* __restrict__ in,
                                                         float* __restrict__ out)
{
    const int b   = blockIdx.x;
    const int tid = threadIdx.x;

    const float* A0 = in  + (size_t)b * N * N;
    float*       O  = out + (size_t)b * N * N;

    // ---- dynamic LDS partition ----
    float* sA   = smem;                       // 128*129
    float* sV   = sA + N * LDA;               // 128*129
    float* sC   = sV + N * LDA;               // 64 rotation cosines
    float* sS   = sC + NPAIR;                 // 64 rotation sines
    int*   sTop = (int*)(sS + NPAIR);         // 64 pair indices
    int*   sBot = sTop + NPAIR;               // 64 pair indices
    float* sLw  = (float*)(sBot + NPAIR);     // 128 log-eigenvalues

    // ---- load A (+eps*I), V = I ----
    for (int idx = tid; idx < N * N; idx += BLOCK) {
        int i = idx >> 7, j = idx & 127;
        float v = A0[idx];
        if (i == j) v += EPS;
        sA[i * LDA + j] = v;
        sV[i * LDA + j] = (i == j) ? 1.0f : 0.0f;
    }
    if (tid < NPAIR) { sTop[tid] = 2 * tid; sBot[tid] = 2 * tid + 1; }
    __syncthreads();

    // ---- parallel cyclic Jacobi ----
    for (int sweep = 0; sweep < NSWEEPS; ++sweep) {
        for (int step = 0; step < N - 1; ++step) {
            // rotation parameters: one thread per pair
            if (tid < NPAIR) {
                int p = sTop[tid], q = sBot[tid];
                if (p > q) { int t = p; p = q; q = t; }
                float app = sA[p * LDA + p];
                float aqq = sA[q * LDA + q];
                float apq = sA[p * LDA + q];
                float c = 1.0f, sv = 0.0f;
                if (fabsf(apq) > 1e-13f) {
                    float tau = (aqq - app) / (2.0f * apq);
                    float t   = copysignf(1.0f, tau) /
                                (fabsf(tau) + sqrtf(1.0f + tau * tau));
                    c  = 1.0f / sqrtf(1.0f + t * t);
                    sv = t * c;
                }
                sC[tid] = c; sS[tid] = sv;
            }
            __syncthreads();

            // row update: A[p,:], A[q,:].  lane <-> column, 32 pairs per half-block
            {
                int j      = tid & 127;
                int prBase = (tid >> 7) * 32;
                for (int pp = 0; pp < 32; ++pp) {
                    int pr = prBase + pp;
                    int p = sTop[pr], q = sBot[pr];
                    if (p > q) { int t = p; p = q; q = t; }
                    float c = sC[pr], sv = sS[pr];
                    float ap = sA[p * LDA + j], aq = sA[q * LDA + j];
                    sA[p * LDA + j] = c * ap - sv * aq;
                    sA[q * LDA + j] = sv * ap + c * aq;
                }
            }
            __syncthreads();

            // column update on A and V: lane <-> row (stride-129 -> conflict-free)
            {
                int i      = tid & 127;
                int prBase = (tid >> 7) * 32;
                for (int pp = 0; pp < 32; ++pp) {
                    int pr = prBase + pp;
                    int p = sTop[pr], q = sBot[pr];
                    if (p > q) { int t = p; p = q; q = t; }
                    float c = sC[pr], sv = sS[pr];
                    float ap = sA[i * LDA + p], aq = sA[i * LDA + q];
                    sA[i * LDA + p] = c * ap - sv * aq;
                    sA[i * LDA + q] = sv * ap + c * aq;
                    float vp = sV[i * LDA + p], vq = sV[i * LDA + q];
                    sV[i * LDA + p] = c * vp - sv * vq;
                    sV[i * LDA + q] = sv * vp + c * vq;
                }
            }
            __syncthreads();

            // round-robin tournament permutation (serial, tiny)
            if (tid == 0) {
                int newtop1 = sBot[0];
                int lastTop = sTop[NPAIR - 1];
                for (int i = NPAIR - 1; i >= 2; --i) sTop[i] = sTop[i - 1];
                for (int i = 0; i < NPAIR - 1; ++i)  sBot[i] = sBot[i + 1];
                sTop[1] = newtop1;
                sBot[NPAIR - 1] = lastTop;
            }
            __syncthreads();
        }
    }

    // ---- eigenvalues -> rectified log ----
    if (tid < N) {
        float w = sA[tid * LDA + tid];
        sLw[tid] = logf(fmaxf(w, RECT_EPS));
    }
    __syncthreads();

    // W = V * diag(lw)  (reuse sA)
    for (int idx = tid; idx < N * N; idx += BLOCK) {
        int i = idx >> 7, j = idx & 127;
        sA[i * LDA + j] = sV[i * LDA + j] * sLw[j];
    }
    __syncthreads();

    // ---- out = W x V^T via v_wmma_f32_16x16x4_f32 ----
    // A frag (16x4 f32): lane m / m+16 hold row M=m; VGPR0 = K {0|2}, VGPR1 = K {1|3}
    // B frag (4x16 f32): lane n / n+16 hold col N=n; VGPR0 = K {0|2}, VGPR1 = K {1|3}
    // B[k][n] = V[tj*16+n][k]  (i.e. V^T)
    const int wave = tid >> 5;          // 0..7
    const int lane = tid & 31;
    const int r    = lane & 15;
    const int koff = (lane < 16) ? 0 : 2;

    for (int tile = wave; tile < 64; tile += 8) {
        const int ti = tile >> 3, tj = tile & 7;
        const float* wrow = &sA[(ti * 16 + r) * LDA];
        const float* vrow = &sV[(tj * 16 + r) * LDA];
        v8f acc = {};
#pragma unroll 4
        for (int k0 = 0; k0 < N; k0 += 4) {
            v2f a, bb;
            a.x  = wrow[k0 + koff];
            a.y  = wrow[k0 + koff + 1];
            bb.x = vrow[k0 + koff];
            bb.y = vrow[k0 + koff + 1];
            acc = __builtin_amdgcn_wmma_f32_16x16x4_f32(
                /*neg_a=*/false, a, /*neg_b=*/false, bb,
                /*c_mod=*/(short)0, acc, /*reuse_a=*/false, /*reuse_b=*/false);
        }
        // D layout: VGPR rr -> M = rr (+8 for upper lanes), N = lane%16
        const int mbase = ti * 16 + ((lane < 16) ? 0 : 8);
        const int col   = tj * 16 + r;
#pragma unroll
        for (int rr = 0; rr < 8; ++rr) {
            O[(size_t)(mbase + rr) * N + col] = acc[rr];
        }
    }
}

extern "C" void kernel_launch(void* const* d_in, const int* in_sizes, int n_in,
                              void* d_out, int out_size, void* d_ws, size_t ws_size,
                              hipStream_t stream) {
    const float* h = (const float*)d_in[0];
    float* o       = (float*)d_out;
    const int B    = in_sizes[0] / (N * N);

    const size_t shmem =
        (size_t)(2 * N * LDA          // sA + sV
                 + 2 * NPAIR          // sC + sS
                 + 2 * NPAIR          // sTop + sBot (ints, same width)
                 + N                  // sLw
        ) * sizeof(float);            // = 133,632 bytes (fits CDNA5 320KB WGP LDS)

    spd_logm_kernel<<<B, BLOCK, shmem, stream>>>(h, o);
}